// SkipTopNCrossEntropy_64733747085575
// MI455X (gfx1250) — compile-verified
//
#include <hip/hip_runtime.h>

#define CDIM   8192
#define LSM    0.1f
#define NSKIP  3
#define TPB    256
#define PER_T  (CDIM / TPB)     // 32 floats per thread
#define CHUNKS (PER_T / 4)      // 8 float4 (b128) transfers per thread

__device__ __forceinline__ void topk_insert(float (&tv)[4], int (&ti)[4], float x, int c) {
    if (x <= tv[3]) return;
    tv[3] = x; ti[3] = c;
#pragma unroll
    for (int p = 3; p > 0; --p) {
        if (tv[p] > tv[p - 1]) {
            float fv = tv[p]; tv[p] = tv[p - 1]; tv[p - 1] = fv;
            int   iv = ti[p]; ti[p] = ti[p - 1]; ti[p - 1] = iv;
        }
    }
}

// Compile-time-unrolled consumption of async chunks. Async-LDS ops complete
// in order, so waiting asynccnt <= (CHUNKS-1-K) guarantees chunk K has landed
// in LDS for this wave's lanes (each lane reads back only its own slice).
template <int K>
__device__ __forceinline__ void consume_chunks(const float4* __restrict__ rp, int t,
                                               float4 (&vals)[CHUNKS],
                                               float& m, float& T,
                                               float (&tv)[4], int (&ti)[4]) {
    if constexpr (K < CHUNKS) {
        asm volatile("s_wait_asynccnt %0" :: "i"(CHUNKS - 1 - K) : "memory");
        vals[K] = rp[K * TPB + t];
        const int cb = K * (TPB * 4) + t * 4;
        float x;
        x = vals[K].x; m = fmaxf(m, x); T += x; topk_insert(tv, ti, x, cb + 0);
        x = vals[K].y; m = fmaxf(m, x); T += x; topk_insert(tv, ti, x, cb + 1);
        x = vals[K].z; m = fmaxf(m, x); T += x; topk_insert(tv, ti, x, cb + 2);
        x = vals[K].w; m = fmaxf(m, x); T += x; topk_insert(tv, ti, x, cb + 3);
        consume_chunks<K + 1>(rp, t, vals, m, T, tv, ti);
    }
}

__global__ __launch_bounds__(TPB) void skip_topn_ce_row_kernel(
        const float* __restrict__ preds,
        const int*   __restrict__ targets,
        float*       __restrict__ row_loss) {
    __shared__ float rowbuf[CDIM];          // 32 KB: whole row staged via async DMA
    __shared__ float s_m[TPB], s_s[TPB], s_T[TPB];
    __shared__ float s_tv[TPB][4];
    __shared__ int   s_ti[TPB][4];

    const int t   = threadIdx.x;
    const int row = blockIdx.x;
    const char* gbase = (const char*)(preds + (size_t)row * CDIM);

    // ---- CDNA5 async global->LDS streaming (ASYNCcnt path), non-temporal ----
    unsigned lds0 = (unsigned)(size_t)(void*)rowbuf;   // low 32 bits = LDS offset
#pragma unroll
    for (int k = 0; k < CHUNKS; ++k) {
        unsigned    loff = lds0 + (unsigned)(k * (TPB * 16) + t * 16);
        const void* ga   = (const void*)(gbase + k * (TPB * 16) + t * 16);
        asm volatile("global_load_async_to_lds_b128 %0, %1, off th:TH_LOAD_NT"
                     :: "v"(loff), "v"(ga) : "memory");
    }

    // ---- per-thread reduction, pipelined against the in-flight DMA ----
    const float4* rp = reinterpret_cast<const float4*>(rowbuf);
    float4 vals[CHUNKS];

    float m = -3.4028235e38f, T = 0.f;
    float tv[4] = {-3.4028235e38f, -3.4028235e38f, -3.4028235e38f, -3.4028235e38f};
    int   ti[4] = {-1, -1, -1, -1};

    consume_chunks<0>(rp, t, vals, m, T, tv, ti);

    float s = 0.f;
#pragma unroll
    for (int k = 0; k < CHUNKS; ++k) {
        s += __expf(vals[k].x - m);
        s += __expf(vals[k].y - m);
        s += __expf(vals[k].z - m);
        s += __expf(vals[k].w - m);
    }

    // ---- cross-thread tree merge (online-softmax combine + top-4 merge) ----
    // asynccnt reached 0 in the last consume step, so after this barrier the
    // full row in LDS is visible to all waves (needed for rowbuf[gt]).
    s_m[t] = m; s_s[t] = s; s_T[t] = T;
#pragma unroll
    for (int j = 0; j < 4; ++j) { s_tv[t][j] = tv[j]; s_ti[t][j] = ti[j]; }
    __syncthreads();

    for (int stride = TPB / 2; stride > 0; stride >>= 1) {
        if (t < stride) {
            const int o  = t + stride;
            const float m2 = s_m[o], s2 = s_s[o];
            const float mm = fmaxf(m, m2);
            s = s * __expf(m - mm) + s2 * __expf(m2 - mm);
            m = mm;
            T += s_T[o];
#pragma unroll
            for (int j = 0; j < 4; ++j) topk_insert(tv, ti, s_tv[o][j], s_ti[o][j]);
            s_m[t] = m; s_s[t] = s; s_T[t] = T;
#pragma unroll
            for (int j = 0; j < 4; ++j) { s_tv[t][j] = tv[j]; s_ti[t][j] = ti[j]; }
        }
        __syncthreads();
    }

    if (t == 0) {
        const float logZ   = m + __logf(s);
        const int   gt     = targets[row];
        const float x_gt   = rowbuf[gt];
        const float lp_gt  = x_gt - logZ;
        const float sum_lp = T - (float)CDIM * logZ;

        // sum of logits of the top-3 NON-gt entries (top-4 holds at most one gt)
        float skip_logit = 0.f; int cnt = 0;
#pragma unroll
        for (int j = 0; j < 4; ++j) {
            if (ti[j] != gt && cnt < NSKIP) { skip_logit += tv[j]; ++cnt; }
        }
        const float skip_lp = skip_logit - (float)NSKIP * logZ;
        const float lsv     = LSM / (float)(CDIM - 1);
        row_loss[row] = -(1.0f - LSM) * lp_gt - lsv * (sum_lp - lp_gt - skip_lp);
    }
}

__global__ __launch_bounds__(TPB) void mean_kernel(const float* __restrict__ v,
                                                   float* __restrict__ out) {
    __shared__ float sm[TPB];
    float acc = 0.f;
    for (int i = threadIdx.x; i < CDIM; i += TPB) acc += v[i];
    sm[threadIdx.x] = acc;
    __syncthreads();
    for (int stride = TPB / 2; stride > 0; stride >>= 1) {
        if (threadIdx.x < stride) sm[threadIdx.x] += sm[threadIdx.x + stride];
        __syncthreads();
    }
    if (threadIdx.x == 0) out[0] = sm[0] / (float)CDIM;
}

extern "C" void kernel_launch(void* const* d_in, const int* in_sizes, int n_in,
                              void* d_out, int out_size, void* d_ws, size_t ws_size,
                              hipStream_t stream) {
    (void)in_sizes; (void)n_in; (void)out_size; (void)ws_size;
    const float* preds   = (const float*)d_in[0];
    const int*   targets = (const int*)d_in[1];
    float*       out     = (float*)d_out;
    float*       ws      = (float*)d_ws;   // 8192 row losses

    skip_topn_ce_row_kernel<<<CDIM, TPB, 0, stream>>>(preds, targets, ws);
    mean_kernel<<<1, TPB, 0, stream>>>(ws, out);
}